// DAGExecutor_64862596104271
// MI455X (gfx1250) — compile-verified
//
#include <hip/hip_runtime.h>

// ---------------- problem constants (match reference) ----------------
#define DAG_DEPTH       8
#define NUM_INITIAL     9
#define TOTAL_NODES     17
#define D_TOTAL         8
#define BASEN           10
#define LOGITS_PER_NODE (D_TOTAL * BASEN)                 // 80
#define LOGITS_PER_ELEM (NUM_INITIAL * LOGITS_PER_NODE)   // 720
#define O_PER_ELEM      (DAG_DEPTH * TOTAL_NODES)         // 136

#define LOG_LIM   100.0f
#define MAG_MAX   1e28f
#define MAG_MIN   1e-12f
#define INV_TEMP  100.0f      // 1/TEMP
#define INV_STEMP 1e4f        // 1/SIGN_TEMP

#define K2_BLOCK  64          // 2 waves (wave32) per workgroup
#define K1_BLOCK  256         // 8 waves
#define K1_GRPS   4           // 16-row groups per wave
#define NCHUNK    (LOGITS_PER_NODE / 4)   // 20 WMMA K-chunks of 4

typedef unsigned int u32x4 __attribute__((ext_vector_type(4)));
typedef int          i32x8 __attribute__((ext_vector_type(8)));
typedef int          i32x4 __attribute__((ext_vector_type(4)));
typedef float        v2f   __attribute__((ext_vector_type(2)));
typedef float        v8f   __attribute__((ext_vector_type(8)));

__device__ __forceinline__ float clampf(float x, float lo, float hi) {
    return fminf(fmaxf(x, lo), hi);
}

__device__ __forceinline__ float swz(float v, int offset_imm) { return v; } // unused on host

#if defined(__gfx1250__)
#define SWZ(v, off) __int_as_float(__builtin_amdgcn_ds_swizzle(__float_as_int(v), (off)))
#endif

// Bc[k][n] for the constant 80x16 B matrix; k is compile-time.
__device__ __forceinline__ float bval_k(int k, int n) {
    const int s = k / BASEN, d = k % BASEN;
    return (n == 2 * s) ? (float)d : ((n == 2 * s + 1) ? 1.0f : 0.0f);
}

// Scalar softmax magnitude for one node (fallback path).
__device__ __forceinline__ float node_vmag(const float* __restrict__ p) {
    const float powers[D_TOTAL] = {1e3f, 1e2f, 1e1f, 1e0f, 1e-1f, 1e-2f, 1e-3f, 1e-4f};
    float vmag = 0.0f;
#pragma unroll
    for (int d = 0; d < D_TOTAL; ++d) {
        float s[BASEN];
        const float2* p2 = (const float2*)(p + d * BASEN);
#pragma unroll
        for (int i = 0; i < BASEN / 2; ++i) {
            float2 q = p2[i];
            s[2 * i] = q.x;
            s[2 * i + 1] = q.y;
        }
        float m = s[0];
#pragma unroll
        for (int j = 1; j < BASEN; ++j) m = fmaxf(m, s[j]);
        float den = 0.0f, num = 0.0f;
#pragma unroll
        for (int j = 0; j < BASEN; ++j) {
            float e = __expf((s[j] - m) * INV_TEMP);
            den += e;
            num = fmaf(e, (float)j, num);
        }
        vmag = fmaf(powers[d], num / den, vmag);
    }
    return clampf(vmag, MAG_MIN, MAG_MAX);
}

// Kernel 1: init magnitudes via WMMA. Each wave handles K1_GRPS groups of 16
// rows (row = element*9+node). Row M is held by lane pair (M, M+16); half h
// owns K = 4t+2h, 4t+2h+1. D = E(16x80) x Bc(80x16) via 20x v_wmma_f32_16x16x4_f32.
__global__ __launch_bounds__(K1_BLOCK) void k_init_mag_wmma(const float* __restrict__ logits,
                                                            float* __restrict__ init_mag,
                                                            int n_tasks, int n_groups) {
#if defined(__gfx1250__)
    const int lane = (int)(threadIdx.x & 31u);
    const int n    = lane & 15;          // column / row-in-group
    const int h    = lane >> 4;          // K-half select
    const bool hs  = (h != 0);
    const int gwave = (int)blockIdx.x * (K1_BLOCK / 32) + (int)(threadIdx.x >> 5);

    // ---- constant B fragments (per ISA B layout: vgpr0 rows 4t+2h..) ----
    v2f bf[NCHUNK];
#pragma unroll
    for (int t = 0; t < NCHUNK; ++t) {
        float b0 = bval_k(4 * t + 0, n), b2 = bval_k(4 * t + 2, n);
        float b1 = bval_k(4 * t + 1, n), b3 = bval_k(4 * t + 3, n);
        bf[t].x = hs ? b2 : b0;
        bf[t].y = hs ? b3 : b1;
    }
    // powers[n>>1] = 10^(3-s) via bit selects (no indexed table)
    const int sN = n >> 1;
    const float pw = 1000.0f * ((sN & 1) ? 0.1f : 1.0f)
                             * ((sN & 2) ? 0.01f : 1.0f)
                             * ((sN & 4) ? 1e-4f : 1.0f);

    for (int gi = 0; gi < K1_GRPS; ++gi) {
        const int grp = gwave * K1_GRPS + gi;
        if (grp >= n_groups) break;                  // wave-uniform
        const int base = grp * 16;
        const int row  = min(base + n, n_tasks - 1); // clamp tail rows (loads only)
        const float* rowp = logits + (size_t)row * LOGITS_PER_NODE;

        // ---- load this lane's 40 logits (K = 4t+2h, 4t+2h+1) ----
        float raw[2 * NCHUNK];
#pragma unroll
        for (int t = 0; t < NCHUNK; ++t) {
            float2 q = *(const float2*)(rowp + 4 * t + 2 * h);
            raw[2 * t] = q.x;
            raw[2 * t + 1] = q.y;
        }
        // ---- per-slot max (local half, then combine with partner lane) ----
        float m[D_TOTAL];
#pragma unroll
        for (int s = 0; s < D_TOTAL; ++s) m[s] = -3.0e38f;
#pragma unroll
        for (int j = 0; j < 2 * NCHUNK; ++j) {
            const int K0 = 4 * (j >> 1) + (j & 1);   // K for h=0; h=1 -> K0+2
            const int s0 = K0 / BASEN, s1 = (K0 + 2) / BASEN;
            const float v = raw[j];
            if (s0 == s1) {
                m[s0] = fmaxf(m[s0], v);
            } else {
                float t0 = fmaxf(m[s0], v), t1 = fmaxf(m[s1], v);
                m[s0] = hs ? m[s0] : t0;
                m[s1] = hs ? t1 : m[s1];
            }
        }
#pragma unroll
        for (int s = 0; s < D_TOTAL; ++s)
            m[s] = fmaxf(m[s], SWZ(m[s], 0x401F));   // SWAPX16: partner half
        // ---- exponentials (slot-shift cancels in num/den ratio) ----
        float ex[2 * NCHUNK];
#pragma unroll
        for (int j = 0; j < 2 * NCHUNK; ++j) {
            const int K0 = 4 * (j >> 1) + (j & 1);
            const int s0 = K0 / BASEN, s1 = (K0 + 2) / BASEN;
            const float ms = (s0 == s1) ? m[s0] : (hs ? m[s1] : m[s0]);
            ex[j] = __expf((raw[j] - ms) * INV_TEMP);
        }
        // ---- 20 chained WMMAs: D += E_chunk(16x4) x Bc_chunk(4x16) ----
        v8f acc = {0.0f, 0.0f, 0.0f, 0.0f, 0.0f, 0.0f, 0.0f, 0.0f};
#pragma unroll
        for (int t = 0; t < NCHUNK; ++t) {
            v2f a;
            a.x = ex[2 * t];
            a.y = ex[2 * t + 1];
            acc = __builtin_amdgcn_wmma_f32_16x16x4_f32(false, a, false, bf[t],
                                                        (short)0, acc, false, false);
        }
        // ---- per-row result: sum_s pw[s]*num/den over even columns ----
        float q[8];
#pragma unroll
        for (int r = 0; r < 8; ++r) {
            float den = SWZ(acc[r], 0x041F);         // xor 1: neighbor column
            float val = pw * acc[r] / den;           // valid on even columns
            val = (n & 1) ? 0.0f : val;
            val += SWZ(val, 0x081F);                 // xor 2
            val += SWZ(val, 0x101F);                 // xor 4
            val += SWZ(val, 0x201F);                 // xor 8 -> lane n==0 has sum
            q[r] = clampf(val, MAG_MIN, MAG_MAX);
        }
        if (n == 0) {                                 // lanes 0 and 16 store 8 rows each
            const int sb = base + 8 * h;
#pragma unroll
            for (int r = 0; r < 8; ++r) {
                const int idx = sb + r;
                if (idx < n_tasks) init_mag[idx] = q[r];
            }
        }
    }
#else
    for (int r = (int)(blockIdx.x * blockDim.x + threadIdx.x); r < n_tasks;
         r += (int)(gridDim.x * blockDim.x))
        init_mag[r] = node_vmag(logits + (size_t)r * LOGITS_PER_NODE);
#endif
}

// Kernel 2: one thread per element; O tile staged to LDS via the Tensor Data
// Mover (TENSORcnt), then the 8-step DAG recurrence fully unrolled in registers.
__global__ __launch_bounds__(K2_BLOCK) void k_dag(const float* __restrict__ logits,
                                                  const float* __restrict__ init_mag,
                                                  const float* __restrict__ V_sign,
                                                  const float* __restrict__ O,
                                                  const float* __restrict__ G,
                                                  float* __restrict__ out,
                                                  int n_elem, int use_ws) {
    __shared__ float ldsO[K2_BLOCK * O_PER_ELEM];   // 34 KB per workgroup

    const int e0 = blockIdx.x * K2_BLOCK;
    const int e  = e0 + (int)threadIdx.x;
    const int elems_here = min(K2_BLOCK, n_elem - e0);
    const unsigned ncopy = (unsigned)(elems_here * O_PER_ELEM);  // <= 8704

#if defined(__gfx1250__)
    if (threadIdx.x < 32) {   // wave 0 issues the TDM load
        unsigned long long gaddr =
            (unsigned long long)(const void*)(O + (size_t)e0 * O_PER_ELEM);
        unsigned lds_off = (unsigned)(unsigned long long)(const void*)&ldsO[0];

        u32x4 g0;
        g0[0] = 1u;                                   // count=1
        g0[1] = lds_off;                              // lds_addr
        g0[2] = (unsigned)(gaddr & 0xFFFFFFFFull);    // global_addr lo
        g0[3] = (unsigned)((gaddr >> 32) & 0x1FFFFFFull) | 0x80000000u; // hi | type=2

        i32x8 g1;
        g1[0] = (int)(2u << 16);                          // data_size=4B
        g1[1] = (int)((ncopy & 0xFFFFu) << 16);           // tensor_dim0 lo16
        g1[2] = (int)(((ncopy >> 16) & 0xFFFFu) | (1u << 16)); // tdim0 hi | tdim1=1
        g1[3] = (int)((ncopy & 0xFFFFu) << 16);           // tile_dim0=ncopy
        g1[4] = (int)1u;                                  // tile_dim1=1, tile_dim2=0
        g1[5] = (int)ncopy;                               // stride0 lo32
        g1[6] = 0;
        g1[7] = 0;
        i32x4 g2 = {0, 0, 0, 0};
        i32x4 g3 = {0, 0, 0, 0};
        i32x8 g4 = {0, 0, 0, 0, 0, 0, 0, 0};

        __builtin_amdgcn_tensor_load_to_lds(g0, g1, g2, g3, g4, 0);
        __builtin_amdgcn_s_wait_tensorcnt(0);
    }
    __syncthreads();
#else
    for (int i = threadIdx.x; i < (int)ncopy; i += K2_BLOCK)
        ldsO[i] = O[(size_t)e0 * O_PER_ELEM + i];
    __syncthreads();
#endif

    if (e >= n_elem) return;

    float wmag[TOTAL_NODES], wsign[TOTAL_NODES], lm[TOTAL_NODES];

    if (use_ws) {
#pragma unroll
        for (int i = 0; i < NUM_INITIAL; ++i)
            wmag[i] = init_mag[(size_t)e * NUM_INITIAL + i];
    } else {
        const float* p = logits + (size_t)e * LOGITS_PER_ELEM;
#pragma unroll
        for (int i = 0; i < NUM_INITIAL; ++i)
            wmag[i] = node_vmag(p + i * LOGITS_PER_NODE);
    }
#pragma unroll
    for (int i = NUM_INITIAL; i < TOTAL_NODES; ++i) wmag[i] = 0.0f;
#pragma unroll
    for (int i = 0; i < TOTAL_NODES; ++i)
        wsign[i] = V_sign[(size_t)e * TOTAL_NODES + i];
#pragma unroll
    for (int i = 0; i < NUM_INITIAL; ++i)
        lm[i] = __logf(fmaxf(wmag[i], MAG_MIN));

    float gv[DAG_DEPTH];
#pragma unroll
    for (int s = 0; s < DAG_DEPTH; ++s) gv[s] = G[(size_t)e * DAG_DEPTH + s];

    const float* myO = &ldsO[(int)threadIdx.x * O_PER_ELEM];

#pragma unroll
    for (int step = 0; step < DAG_DEPTH; ++step) {
        const int   valid = NUM_INITIAL + step;
        const float g     = gv[step];
        float R = 0.0f, prod = 1.0f;
#pragma unroll
        for (int nn = 0; nn < TOTAL_NODES; ++nn) {
            if (nn < valid) {
                float o     = myO[step * TOTAL_NODES + nn];
                float mixed = lm[nn] * (1.0f - g) + wsign[nn] * wmag[nn] * g;
                R    = fmaf(o, mixed, R);
                prod *= fmaf(wsign[nn] * fabsf(o), 2.0f, 1.0f);
            }
        }
        float linear_sign = tanhf(R * INV_STEMP);
        float log_sign    = tanhf(prod * INV_STEMP);
        float s_new = clampf(g * linear_sign + (1.0f - g) * log_sign, -1.0f, 1.0f);

        float linear_mag  = fminf(fabsf(R), MAG_MAX);
        float log_mag_res = __expf(clampf(R, -LOG_LIM, LOG_LIM));
        float m_new = clampf(g * linear_mag + (1.0f - g) * log_mag_res, MAG_MIN, MAG_MAX);

        wmag[valid]  = m_new;
        wsign[valid] = s_new;
        lm[valid]    = __logf(fmaxf(m_new, MAG_MIN));
    }

    out[e] = wsign[TOTAL_NODES - 1] * wmag[TOTAL_NODES - 1];
}

extern "C" void kernel_launch(void* const* d_in, const int* in_sizes, int n_in,
                              void* d_out, int out_size, void* d_ws, size_t ws_size,
                              hipStream_t stream) {
    const float* logits = (const float*)d_in[0];   // [B,T,9,8,10]
    const float* V_sign = (const float*)d_in[1];   // [B,T,17]
    const float* O      = (const float*)d_in[2];   // [B,T,8,17]
    const float* G      = (const float*)d_in[3];   // [B,T,8]
    float* out = (float*)d_out;                    // [B,T]

    const int n_elem  = out_size;                  // B*T
    const int n_tasks = n_elem * NUM_INITIAL;
    const size_t need = (size_t)n_tasks * sizeof(float);
    const int use_ws  = (ws_size >= need) ? 1 : 0;
    float* init_mag   = (float*)d_ws;

    if (use_ws) {
        const int n_groups = (n_tasks + 15) / 16;
        const int n_waves  = (n_groups + K1_GRPS - 1) / K1_GRPS;
        const int n_blocks = (n_waves + (K1_BLOCK / 32) - 1) / (K1_BLOCK / 32);
        k_init_mag_wmma<<<n_blocks, K1_BLOCK, 0, stream>>>(logits, init_mag,
                                                           n_tasks, n_groups);
    }
    const int nblk = (n_elem + K2_BLOCK - 1) / K2_BLOCK;
    k_dag<<<nblk, K2_BLOCK, 0, stream>>>(logits, init_mag, V_sign, O, G, out,
                                         n_elem, use_ws);
}